// DGCNN_82738249990500
// MI455X (gfx1250) — compile-verified
//
#include <hip/hip_runtime.h>

// ---------------------------------------------------------------------------
// DGCNN head on MI455X (gfx1250, wave32).
//  * Convs = GEMMs on V_WMMA_F32_16X16X4_F32 (fp32 reference numerics kept).
//  * Activations stored position-major (P, C) so every WMMA operand is one
//    global_load_b64 and every epilogue store is b128.
//  * 2 N-tiles per wave: each A load feeds two WMMAs.
//  * BN statistics fused into the GEMM epilogue (shfl_xor + atomics).
//  * KNN point staging uses CDNA5 async global->LDS loads.
// ---------------------------------------------------------------------------

typedef __attribute__((ext_vector_type(2))) float v2f;
typedef __attribute__((ext_vector_type(8))) float v8f;

#define B_DIM 8
#define T_DIM 3
#define N_PTS 2048
#define KNN_K 3
#define W_DIM (N_PTS * KNN_K)          // 6144
#define TW (T_DIM * W_DIM)             // 18432 positions per batch item
#define NCOLS (B_DIM * TW)             // 147456 GEMM columns
#define CMAX 64
#define BIGF 3.402823466e38f

__device__ __forceinline__ float4 fmax4(float4 a, float4 b) {
    float4 r;
    r.x = fmaxf(a.x, b.x); r.y = fmaxf(a.y, b.y);
    r.z = fmaxf(a.z, b.z); r.w = fmaxf(a.w, b.w);
    return r;
}

// ---------------------------------------------------------------------------
// Kernel 1: KNN + edge-feature build -> xin (P, 4) position-major.
// grid = 24 frames * 8 segments; block = 256 threads, one point per thread.
// Point set staged to LDS with CDNA5 async global->LDS loads (ASYNCcnt).
// Replicates jax.lax.top_k(-dist, 4)[1:]: ascending (dist, idx), drop first.
// ---------------------------------------------------------------------------
__global__ void knn_build_kernel(const float* __restrict__ lm,   // (B,2,T,N)
                                 float* __restrict__ xin)        // (P,4)
{
    const int bt  = blockIdx.x >> 3;          // 0..23
    const int seg = blockIdx.x & 7;
    const int b = bt / T_DIM, t = bt % T_DIM;

    __shared__ float px[N_PTS], py[N_PTS], ps[N_PTS];
    const float* bx = lm + ((size_t)(b * 2 + 0) * T_DIM + t) * N_PTS;
    const float* by = lm + ((size_t)(b * 2 + 1) * T_DIM + t) * N_PTS;

    // Async stage: each thread pulls its own slots straight into LDS.
    for (int i = threadIdx.x; i < N_PTS; i += 256) {
        unsigned lx = (unsigned)(uintptr_t)(&px[i]);
        unsigned ly = (unsigned)(uintptr_t)(&py[i]);
        const float* gx = bx + i;
        const float* gy = by + i;
        asm volatile("global_load_async_to_lds_b32 %0, %1, off"
                     :: "v"(lx), "v"(gx) : "memory");
        asm volatile("global_load_async_to_lds_b32 %0, %1, off"
                     :: "v"(ly), "v"(gy) : "memory");
    }
    asm volatile("s_wait_asynccnt 0x0" ::: "memory");
    // Each thread squared-norms exactly the slots it loaded (no cross-wave dep).
    for (int i = threadIdx.x; i < N_PTS; i += 256) {
        float x = px[i], y = py[i];
        ps[i] = x * x + y * y;
    }
    __syncthreads();

    const int n = seg * 256 + threadIdx.x;    // 0..2047
    const float xn = px[n], yn = py[n], sn = ps[n];

    float d0 = BIGF, d1 = BIGF, d2 = BIGF, d3 = BIGF;
    int   i0 = 0,    i1 = 0,    i2 = 0,    i3 = 0;
    for (int m = 0; m < N_PTS; ++m) {
        float d = sn + ps[m] - 2.0f * (xn * px[m] + yn * py[m]);
        if (d < d3) {                          // strict < => ties keep lower idx
            d3 = d; i3 = m;
            if (d3 < d2) { float td=d3; d3=d2; d2=td; int ti=i3; i3=i2; i2=ti;
              if (d2 < d1) { td=d2; d2=d1; d1=td; ti=i2; i2=i1; i1=ti;
                if (d1 < d0) { td=d1; d1=d0; d0=td; ti=i1; i1=i0; i0=ti; } } }
        }
    }
    // drop element 0 (self / closest) -> neighbors i1,i2,i3
    const size_t p0 = (size_t)(b * T_DIM + t) * W_DIM + (size_t)n * KNN_K;
    int nb0 = i1, nb1 = i2, nb2 = i3;
    #pragma unroll
    for (int kk = 0; kk < KNN_K; ++kk) {
        int m = (kk == 0) ? nb0 : (kk == 1) ? nb1 : nb2;
        float4 e;
        e.x = xn;            // ch0: x1.x
        e.y = yn;            // ch1: x1.y
        e.z = px[m] - xn;    // ch2: (knn-x1).x
        e.w = py[m] - yn;    // ch3: (knn-x1).y
        *(float4*)(xin + (p0 + kk) * 4) = e;   // one b128 store
    }
}

// ---------------------------------------------------------------------------
// Kernel 2: conv-as-GEMM, WMMA f32 16x16x4, fused BN statistics.
// Activations position-major (P, Cin) -> each lane's K-pair is one b64 load.
// Weights pre-repacked to (taps, Cout, Cin) -> A K-pair is one b64 load.
// One wave = 16(M) x 32(N): two accumulators share every A load.
// W_DIM % 32 == 0 => wave-uniform t => EXEC all-ones at every WMMA.
// ---------------------------------------------------------------------------
__global__ void conv_gemm_wmma_kernel(const float* __restrict__ in,   // (P,Cin)
                                      const float* __restrict__ wgtT, // (taps,Cout,Cin)
                                      const float* __restrict__ bias, // (Cout)
                                      float* __restrict__ out,        // (P,Cout)
                                      float* __restrict__ csum,       // (Cout)
                                      float* __restrict__ csumsq,     // (Cout)
                                      int Cin, int Cout, int taps)
{
    const int lane  = threadIdx.x & 31;
    const int wave  = threadIdx.x >> 5;
    const int l15   = lane & 15;
    const int lhalf = lane >> 4;               // K half / M half

    const int mtiles = Cout >> 4;              // 4
    const int tileId = blockIdx.x * 8 + wave;
    const int ntile  = tileId / mtiles;        // 4 waves share the same columns
    const int mtile  = tileId - ntile * mtiles;

    const int col0  = ntile * 32;              // 32 columns per wave
    const int mbase = mtile * 16;
    const int b     = col0 / TW;
    const int r     = col0 - b * TW;           // t*W + w0
    const int t     = r / W_DIM;
    const int w0    = r - t * W_DIM;

    v8f acc0 = {0.f,0.f,0.f,0.f,0.f,0.f,0.f,0.f};
    v8f acc1 = {0.f,0.f,0.f,0.f,0.f,0.f,0.f,0.f};

    const int arow = mbase + l15;              // A row = output channel
    for (int dt = 0; dt < taps; ++dt) {
        const int tp = (taps == 1) ? t : (t + dt - 1);
        if (tp < 0 || tp >= T_DIM) continue;   // zero-pad tap (wave-uniform)
        const float* brow = in + ((size_t)(b * T_DIM + tp) * W_DIM + w0 + l15) * Cin;
        const float* awt  = wgtT + ((size_t)dt * Cout + arow) * Cin;
        #pragma unroll 4
        for (int k = 0; k < Cin; k += 4) {
            const int ka = k + 2 * lhalf;      // this half-wave's K pair
            v2f a  = *(const v2f*)(awt + ka);
            v2f b0 = *(const v2f*)(brow + ka);
            v2f b1 = *(const v2f*)(brow + (size_t)16 * Cin + ka);
            acc0 = __builtin_amdgcn_wmma_f32_16x16x4_f32(
                       false, a, false, b0, (short)0, acc0, false, false);
            acc1 = __builtin_amdgcn_wmma_f32_16x16x4_f32(
                       false, a, false, b1, (short)0, acc1, false, false);
        }
    }

    // Epilogue: bias, b128 stores, fused BN partials (one atomic/channel/wave).
    const int chbase = mbase + 8 * lhalf;      // acc element v -> channel chbase+v
    const float4 bb0 = *(const float4*)(bias + chbase);
    const float4 bb1 = *(const float4*)(bias + chbase + 4);
    float val0[8], val1[8];
    #pragma unroll
    for (int v = 0; v < 8; ++v) {
        const float bv = (v < 4) ? ((const float*)&bb0)[v] : ((const float*)&bb1)[v - 4];
        val0[v] = acc0[v] + bv;
        val1[v] = acc1[v] + bv;
    }
    float* o0 = out + (size_t)(col0 + l15) * Cout + chbase;
    *(float4*)(o0)                        = make_float4(val0[0], val0[1], val0[2], val0[3]);
    *(float4*)(o0 + 4)                    = make_float4(val0[4], val0[5], val0[6], val0[7]);
    *(float4*)(o0 + (size_t)16 * Cout)    = make_float4(val1[0], val1[1], val1[2], val1[3]);
    *(float4*)(o0 + (size_t)16 * Cout + 4)= make_float4(val1[4], val1[5], val1[6], val1[7]);

    #pragma unroll
    for (int v = 0; v < 8; ++v) {
        float s = val0[v] + val1[v];
        float q = val0[v] * val0[v] + val1[v] * val1[v];
        #pragma unroll
        for (int m = 1; m < 16; m <<= 1) {     // stays inside the 16-lane half
            s += __shfl_xor(s, m);
            q += __shfl_xor(q, m);
        }
        if (l15 == 0) {
            atomicAdd(&csum[chbase + v], s);
            atomicAdd(&csumsq[chbase + v], q);
        }
    }
}

// ---------------------------------------------------------------------------
// Repack (Cout,Cin,3,1) conv weights -> (3, Cout, Cin) so A K-pairs are
// contiguous b64 loads.
// ---------------------------------------------------------------------------
__global__ void repack_w3_kernel(const float* __restrict__ w,
                                 float* __restrict__ wT)
{
    const int i = blockIdx.x * blockDim.x + threadIdx.x;
    if (i >= 64 * 64 * 3) return;
    const int dt = i % 3;
    const int rem = i / 3;
    const int ii = rem % 64;
    const int o  = rem / 64;
    wT[((size_t)dt * 64 + o) * 64 + ii] = w[i];
}

// ---------------------------------------------------------------------------
// Kernel 3: BN + ReLU, float4-vectorized (4 consecutive channels per thread).
// ---------------------------------------------------------------------------
__global__ void bn_relu_kernel(const float* __restrict__ y,
                               const float* __restrict__ csum,
                               const float* __restrict__ csumsq,
                               const float* __restrict__ gamma,
                               const float* __restrict__ beta,
                               float* __restrict__ z, int Cout)
{
    const size_t i = (size_t)blockIdx.x * blockDim.x + threadIdx.x; // /4 index
    const int ch0 = (int)((i * 4) % Cout);
    const float cnt = (float)NCOLS;
    float4 v = *(const float4*)(y + i * 4);
    float* vp = (float*)&v;
    #pragma unroll
    for (int c = 0; c < 4; ++c) {
        const int ch = ch0 + c;
        const float m   = csum[ch] / cnt;
        const float var = fmaxf(csumsq[ch] / cnt - m * m, 0.0f);
        const float inv = rsqrtf(var + 1e-5f);
        vp[c] = fmaxf((vp[c] - m) * inv * gamma[ch] + beta[ch], 0.0f);
    }
    *(float4*)(z + i * 4) = v;
}

// ---------------------------------------------------------------------------
// Kernel 4: feat[b,c] = max over positions. h is (P, 64) position-major.
// grid = 8 * 18; block 256 = 16 rows x 16 channel-groups; coalesced float4.
// Combine across blocks with uint atomicMax (valid: post-ReLU >= 0, feat=0).
// ---------------------------------------------------------------------------
__global__ void feat_max_kernel(const float* __restrict__ h,
                                float* __restrict__ feat)
{
    const int b   = blockIdx.x / 18;
    const int seg = blockIdx.x % 18;
    const int g   = threadIdx.x & 15;          // channel group: ch = 4g..4g+3
    const int rr  = threadIdx.x >> 4;          // 0..15
    float4 mx = make_float4(0.f, 0.f, 0.f, 0.f);
    #pragma unroll 4
    for (int j = 0; j < 64; ++j) {
        const size_t p = (size_t)b * TW + seg * 1024 + rr + 16 * j;
        mx = fmax4(mx, *(const float4*)(h + p * 64 + g * 4));
    }
    __shared__ float4 red[256];
    red[threadIdx.x] = mx;
    __syncthreads();
    for (int s = 128; s >= 16; s >>= 1) {
        if (threadIdx.x < s) red[threadIdx.x] = fmax4(red[threadIdx.x], red[threadIdx.x + s]);
        __syncthreads();
    }
    if (threadIdx.x < 16) {
        float4 v = red[threadIdx.x];
        unsigned* f = (unsigned*)(feat + b * 64 + threadIdx.x * 4);
        atomicMax(f + 0, __float_as_uint(v.x));
        atomicMax(f + 1, __float_as_uint(v.y));
        atomicMax(f + 2, __float_as_uint(v.z));
        atomicMax(f + 3, __float_as_uint(v.w));
    }
}

// ---------------------------------------------------------------------------
// Kernel 5: FC head: out[b,j] = feat[b,:] . fc_w[:,j] + fc_b[j]
// ---------------------------------------------------------------------------
__global__ void fc_kernel(const float* __restrict__ feat,
                          const float* __restrict__ fw,   // (64,1000)
                          const float* __restrict__ fb,   // (1000)
                          float* __restrict__ outp)
{
    const int i = blockIdx.x * blockDim.x + threadIdx.x;
    if (i >= B_DIM * 1000) return;
    const int b = i / 1000, j = i - b * 1000;
    float acc = fb[j];
    #pragma unroll 8
    for (int c = 0; c < 64; ++c) acc += feat[b * 64 + c] * fw[c * 1000 + j];
    outp[i] = acc;
}

__global__ void zero_buf_kernel(float* __restrict__ p, int n)
{
    const int i = blockIdx.x * blockDim.x + threadIdx.x;
    if (i < n) p[i] = 0.0f;
}

// ---------------------------------------------------------------------------
// Host launcher.
// d_in: 0 landmark | block i at 1+8i: c1w,c1b,g1,b1,c2w,c2b,g2,b2 | 25 fc_w |
// 26 fc_b.  d_out: (8,1000) f32.
// ws floats: bufA(9437184) | bufB(9437184) | sum(64) | sumsq(64) | feat(512)
//            | wT(12288)
// ---------------------------------------------------------------------------
extern "C" void kernel_launch(void* const* d_in, const int* in_sizes, int n_in,
                              void* d_out, int out_size, void* d_ws, size_t ws_size,
                              hipStream_t stream)
{
    const float* landmark = (const float*)d_in[0];
    const float* fc_w     = (const float*)d_in[25];
    const float* fc_b     = (const float*)d_in[26];

    const size_t BIG = (size_t)NCOLS * CMAX;        // 9437184 floats
    float* bufA  = (float*)d_ws;
    float* bufB  = bufA + BIG;
    float* s_sum = bufB + BIG;
    float* s_sq  = s_sum + 64;
    float* feat  = s_sq + 64;
    float* wT    = feat + 512;

    // 1) KNN + edge features -> bufA (P, 4)
    knn_build_kernel<<<24 * 8, 256, 0, stream>>>(landmark, bufA);
    zero_buf_kernel<<<2, 256, 0, stream>>>(feat, 512);

    // 2) Three conv-BN-ReLU blocks; activations ping-pong bufA <-> bufB.
    const int convGrid = (4 * (NCOLS / 32)) / 8;    // 18432 wave-tiles / 8
    const int ewGrid   = (int)((BIG / 4 + 255) / 256);
    for (int blk = 0; blk < 3; ++blk) {
        const int base = 1 + 8 * blk;
        const float* c1w = (const float*)d_in[base + 0];
        const float* c1b = (const float*)d_in[base + 1];
        const float* g1  = (const float*)d_in[base + 2];
        const float* b1  = (const float*)d_in[base + 3];
        const float* c2w = (const float*)d_in[base + 4];
        const float* c2b = (const float*)d_in[base + 5];
        const float* g2  = (const float*)d_in[base + 6];
        const float* b2  = (const float*)d_in[base + 7];
        const int Cin = (blk == 0) ? 4 : 64;

        zero_buf_kernel<<<1, 128, 0, stream>>>(s_sum, 128);
        conv_gemm_wmma_kernel<<<convGrid, 256, 0, stream>>>(
            bufA, c1w, c1b, bufB, s_sum, s_sq, Cin, 64, 1);      // (1,64,Cin)
        bn_relu_kernel<<<ewGrid, 256, 0, stream>>>(bufB, s_sum, s_sq, g1, b1, bufA, 64);

        repack_w3_kernel<<<48, 256, 0, stream>>>(c2w, wT);       // -> (3,64,64)
        zero_buf_kernel<<<1, 128, 0, stream>>>(s_sum, 128);
        conv_gemm_wmma_kernel<<<convGrid, 256, 0, stream>>>(
            bufA, wT, c2b, bufB, s_sum, s_sq, 64, 64, 3);
        bn_relu_kernel<<<ewGrid, 256, 0, stream>>>(bufB, s_sum, s_sq, g2, b2, bufA, 64);
    }

    // 3) Global max pooling -> feat (8,64)
    feat_max_kernel<<<B_DIM * 18, 256, 0, stream>>>(bufA, feat);

    // 4) FC head -> d_out (8,1000)
    fc_kernel<<<(B_DIM * 1000 + 255) / 256, 256, 0, stream>>>(
        feat, fc_w, fc_b, (float*)d_out);
}